// TripletAnchorLoss_85796266705605
// MI455X (gfx1250) — compile-verified
//
#include <hip/hip_runtime.h>

typedef __attribute__((ext_vector_type(2))) float v2f;
typedef __attribute__((ext_vector_type(8))) float v8f;

#define D_DIM 1024
#define N_TRIPLETS 8192
#define TRIPLETS_PER_TILE 16
#define WAVES_PER_BLOCK 4
#define K_PER_WAVE (D_DIM / WAVES_PER_BLOCK) /* 256 */
#define STEPS (K_PER_WAVE / 4)               /* 64  */

__global__ void zero_out_kernel(float* out) { *out = 0.0f; }

__global__ __launch_bounds__(WAVES_PER_BLOCK * 32)
void triplet_wmma_kernel(const float* __restrict__ z,
                         const int* __restrict__ a_idx,
                         const int* __restrict__ p_idx,
                         const int* __restrict__ n_idx,
                         float* __restrict__ out)
{
    const int tid  = threadIdx.x;
    const int lane = tid & 31;
    const int wave = tid >> 5;
    const int m    = lane & 15;  // triplet-in-tile served by this lane
    const int h    = lane >> 4;  // K-half select (A/B layout: lanes 16-31 hold K+2,K+3)
    const int t    = blockIdx.x * TRIPLETS_PER_TILE + m;

    const float* __restrict__ pa = z + (size_t)a_idx[t] * D_DIM;
    const float* __restrict__ pp = z + (size_t)p_idx[t] * D_DIM;
    const float* __restrict__ pn = z + (size_t)n_idx[t] * D_DIM;

    const int kBase = wave * K_PER_WAVE + 2 * h;

    v8f cap = {};  // accumulates diag -> d_ap^2 partials
    v8f can = {};  // accumulates diag -> d_an^2 partials

    // Whole wave executes uniformly: EXEC all-ones as WMMA requires.
    #pragma unroll 4
    for (int s = 0; s < STEPS; ++s) {
        const int col = kBase + 4 * s;
        v2f a = *(const v2f*)(pa + col);
        v2f p = *(const v2f*)(pp + col);
        v2f n = *(const v2f*)(pn + col);
        v2f dp = a - p;
        v2f dn = a - n;
        // D = M * M^T : identical per-lane register image serves as both A and B.
        cap = __builtin_amdgcn_wmma_f32_16x16x4_f32(false, dp, false, dp,
                                                    (short)0, cap, false, false);
        can = __builtin_amdgcn_wmma_f32_16x16x4_f32(false, dn, false, dn,
                                                    (short)0, can, false, false);
    }

    // Extract diagonal: (i,i) lives at {lane=i, vgpr=i} for i<8 and
    // {lane=i+16, vgpr=i-8} for i>=8  =>  vgpr = lane & 7 on lanes 0-7 & 24-31.
    float dap2 = 0.0f, dan2 = 0.0f;
    const int sel = lane & 7;
    #pragma unroll
    for (int v = 0; v < 8; ++v) {
        if (sel == v) { dap2 = cap[v]; dan2 = can[v]; }
    }
    const bool valid = (lane < 8) || (lane >= 24);

    __shared__ float red_ap[WAVES_PER_BLOCK][TRIPLETS_PER_TILE];
    __shared__ float red_an[WAVES_PER_BLOCK][TRIPLETS_PER_TILE];
    if (valid) {
        red_ap[wave][m] = dap2;
        red_an[wave][m] = dan2;
    }
    __syncthreads();

    if (tid < TRIPLETS_PER_TILE) {
        float sap = 0.0f, san = 0.0f;
        #pragma unroll
        for (int w = 0; w < WAVES_PER_BLOCK; ++w) {
            sap += red_ap[w][tid];
            san += red_an[w][tid];
        }
        float hinge = sqrtf(sap) - sqrtf(san) + 1.0f; // MARGIN = 1.0
        hinge = fmaxf(hinge, 0.0f);
        // tree-reduce the 16 hinge values (sources stay within lanes 0..15)
        #pragma unroll
        for (int off = 8; off >= 1; off >>= 1)
            hinge += __shfl_down(hinge, off, 32);
        if (tid == 0)
            atomicAdd(out, hinge * (1.0f / (float)N_TRIPLETS));
    }
}

extern "C" void kernel_launch(void* const* d_in, const int* in_sizes, int n_in,
                              void* d_out, int out_size, void* d_ws, size_t ws_size,
                              hipStream_t stream) {
    const float* z     = (const float*)d_in[0];
    /* d_in[1] = labels (unused by the loss) */
    const int*   a_idx = (const int*)d_in[2];
    const int*   p_idx = (const int*)d_in[3];
    const int*   n_idx = (const int*)d_in[4];
    float*       out   = (float*)d_out;

    zero_out_kernel<<<1, 1, 0, stream>>>(out);
    triplet_wmma_kernel<<<N_TRIPLETS / TRIPLETS_PER_TILE, WAVES_PER_BLOCK * 32, 0, stream>>>(
        z, a_idx, p_idx, n_idx, out);
}